// HGT_88115549045427
// MI455X (gfx1250) — compile-verified
//
#include <hip/hip_runtime.h>
#include <math.h>
#include <stdint.h>

typedef __attribute__((ext_vector_type(2))) float v2f;
typedef __attribute__((ext_vector_type(8))) float v8f;
typedef __attribute__((ext_vector_type(4))) int   v4i;

#if __has_builtin(__builtin_amdgcn_global_load_async_to_lds_b128)
#define HGT_ASYNC_LDS 1
#endif

#define ASG __attribute__((address_space(1)))
#define ASL __attribute__((address_space(3)))

__device__ __forceinline__ void hgt_wait_async() {
#if defined(HGT_ASYNC_LDS)
#if __has_builtin(__builtin_amdgcn_s_wait_asynccnt)
  __builtin_amdgcn_s_wait_asynccnt(0);
#else
  asm volatile("s_wait_asynccnt 0x0" ::: "memory");
#endif
#endif
}

// Copy 16 bytes global -> LDS through the async DMA engine when available,
// else through VGPRs (still exercises ds_store path).
__device__ __forceinline__ void hgt_cp16(const float* __restrict__ g, float* l) {
#if defined(HGT_ASYNC_LDS)
  __builtin_amdgcn_global_load_async_to_lds_b128(
      (ASG v4i*)(uintptr_t)(const void*)g,
      (ASL v4i*)(uintptr_t)(void*)l, 0, 0);
#else
  *(float4*)l = *(const float4*)g;
#endif
}

// ---------------------------------------------------------------------------
// Generic 128-col GEMM:  Y[N x 128] = act(X[N x K] @ W[K x 128] + bias)
// block = 256 threads = 8 wave32; wave w -> column tile w*16; block -> 16 rows.
// Weight panel staged into LDS in TK-row chunks via async global->LDS copies.
// Uses V_WMMA_F32_16X16X4_F32 (full fp32, matches reference precision).
// ---------------------------------------------------------------------------
#define TK 64

__global__ __launch_bounds__(256) void hgt_gemm128(
    const float* __restrict__ X, const float* __restrict__ W,
    const float* __restrict__ bias, float* __restrict__ Y, int K, int act) {
  __shared__ float sW[TK * 128];                 // 32 KB
  const int lane  = threadIdx.x & 31;
  const int wave  = threadIdx.x >> 5;
  const int lrow  = lane & 15;      // A-row / B-col / D-col index
  const int khalf = lane >> 4;      // selects K pair {0,1} vs {2,3}
  const int r0 = blockIdx.x * 16;
  const int c0 = wave * 16;
  const float* xr = X + (size_t)(r0 + lrow) * K;
  v8f acc = {};
  for (int k0 = 0; k0 < K; k0 += TK) {
    __syncthreads();                             // protect sW reuse
    // cooperative stage: TK*128 floats = 2048 float4s; 8 per thread, no divergence
    for (int i = threadIdx.x; i < TK * 128 / 4; i += 256)
      hgt_cp16(W + (size_t)k0 * 128 + (size_t)i * 4, &sW[i * 4]);
    hgt_wait_async();
    __syncthreads();
    for (int k = 0; k < TK; k += 4) {
      v2f a, b;
      a.x = xr[k0 + k + 2 * khalf];
      a.y = xr[k0 + k + 2 * khalf + 1];
      b.x = sW[(k + 2 * khalf) * 128 + c0 + lrow];
      b.y = sW[(k + 2 * khalf + 1) * 128 + c0 + lrow];
      acc = __builtin_amdgcn_wmma_f32_16x16x4_f32(false, a, false, b,
                                                  (short)0, acc, false, false);
    }
  }
  const float bv = bias[c0 + lrow];
  for (int r = 0; r < 8; ++r) {
    float v = acc[r] + bv;
    if (act == 1) v = fmaxf(v, 0.0f);            // relu
    Y[(size_t)(r0 + r + 8 * khalf) * 128 + c0 + lrow] = v;
  }
}

// ---------------------------------------------------------------------------
// Relation transform: Y[n, h*16+e] = sum_d X[n, h*16+d] * R[h, d, e]
// Block-diagonal GEMM: wave w handles head w (16x16 weight, K=16).
// ---------------------------------------------------------------------------
__global__ __launch_bounds__(256) void hgt_rel_transform(
    const float* __restrict__ X, const float* __restrict__ R,
    float* __restrict__ Y) {
  const int lane  = threadIdx.x & 31;
  const int h     = threadIdx.x >> 5;   // head = wave
  const int lrow  = lane & 15;
  const int khalf = lane >> 4;
  const int r0 = blockIdx.x * 16;
  const float* xr = X + (size_t)(r0 + lrow) * 128 + h * 16;
  const float* Rh = R + h * 256;
  v8f acc = {};
  for (int k = 0; k < 16; k += 4) {
    v2f a, b;
    a.x = xr[k + 2 * khalf];
    a.y = xr[k + 2 * khalf + 1];
    b.x = Rh[(k + 2 * khalf) * 16 + lrow];
    b.y = Rh[(k + 2 * khalf + 1) * 16 + lrow];
    acc = __builtin_amdgcn_wmma_f32_16x16x4_f32(false, a, false, b,
                                                (short)0, acc, false, false);
  }
  for (int r = 0; r < 8; ++r)
    Y[(size_t)(r0 + r + 8 * khalf) * 128 + h * 16 + lrow] = acc[r];
}

// ---------------------------------------------------------------------------
// Output projection + gated residual:
// res = g * (gelu(O) @ Wa + ba) + (1-g) * H,  g = sigmoid(skip)
// ---------------------------------------------------------------------------
__device__ __forceinline__ float gelu_exact(float x) {
  return 0.5f * x * (1.0f + erff(x * 0.70710678118654752f));
}

__global__ __launch_bounds__(256) void hgt_outproj(
    const float* __restrict__ O, const float* __restrict__ Wa,
    const float* __restrict__ ba, const float* __restrict__ H,
    const float* __restrict__ skip, float* __restrict__ res) {
  __shared__ float sW[TK * 128];                 // 32 KB, K=128 in two chunks
  const int lane  = threadIdx.x & 31;
  const int wave  = threadIdx.x >> 5;
  const int lrow  = lane & 15;
  const int khalf = lane >> 4;
  const int r0 = blockIdx.x * 16;
  const int c0 = wave * 16;
  const float* xr = O + (size_t)(r0 + lrow) * 128;
  v8f acc = {};
  for (int k0 = 0; k0 < 128; k0 += TK) {
    __syncthreads();
    for (int i = threadIdx.x; i < TK * 128 / 4; i += 256)
      hgt_cp16(Wa + (size_t)k0 * 128 + (size_t)i * 4, &sW[i * 4]);
    hgt_wait_async();
    __syncthreads();
    for (int k = 0; k < TK; k += 4) {
      v2f a, b;
      a.x = gelu_exact(xr[k0 + k + 2 * khalf]);
      a.y = gelu_exact(xr[k0 + k + 2 * khalf + 1]);
      b.x = sW[(k + 2 * khalf) * 128 + c0 + lrow];
      b.y = sW[(k + 2 * khalf + 1) * 128 + c0 + lrow];
      acc = __builtin_amdgcn_wmma_f32_16x16x4_f32(false, a, false, b,
                                                  (short)0, acc, false, false);
    }
  }
  const float g  = 1.0f / (1.0f + expf(-skip[0]));
  const float bv = ba[c0 + lrow];
  for (int r = 0; r < 8; ++r) {
    const size_t row = (size_t)(r0 + r + 8 * khalf);
    const float aval = acc[r] + bv;
    res[row * 128 + c0 + lrow] = g * aval + (1.0f - g) * H[row * 128 + c0 + lrow];
  }
}

// ---------------------------------------------------------------------------
// Edge kernels (memory-bound gather/scatter phase)
// ---------------------------------------------------------------------------
__device__ __forceinline__ void atomicMaxFloat(float* addr, float val) {
  // Seeded with -FLT_MAX; mixed-sign safe int/uint trick.
  if (!(val < 0.0f)) atomicMax((int*)addr, __float_as_int(val));
  else               atomicMin((unsigned int*)addr, __float_as_uint(val));
}

__global__ void hgt_fill(float* __restrict__ p, float v, size_t n) {
  size_t t = (size_t)blockIdx.x * blockDim.x + threadIdx.x;
  if (t < n) p[t] = v;
}

// alpha[e,h] = (q[dst,h,:] . k_rel[src,h,:]) * p_rel[h] / sqrt(16); atomic max per (dst,h)
__global__ void hgt_alpha(const float* __restrict__ q,
                          const float* __restrict__ krel,
                          const int* __restrict__ src, const int* __restrict__ dst,
                          const float* __restrict__ prel,
                          float* __restrict__ alpha, float* __restrict__ mbuf, int E) {
  int t = blockIdx.x * blockDim.x + threadIdx.x;
  if (t >= E * 8) return;
  const int e = t >> 3, h = t & 7;
  const int s = src[e], d = dst[e];
  const float4* qv = (const float4*)(q    + (size_t)d * 128 + h * 16);
  const float4* kv = (const float4*)(krel + (size_t)s * 128 + h * 16);
  float acc = 0.0f;
#pragma unroll
  for (int i = 0; i < 4; ++i) {
    float4 a = qv[i], b = kv[i];
    acc += a.x * b.x + a.y * b.y + a.z * b.z + a.w * b.w;
  }
  const float al = acc * prel[h] * 0.25f;   // 1/sqrt(D), D=16
  alpha[t] = al;
  atomicMaxFloat(&mbuf[(size_t)d * 8 + h], al);
}

// alpha[e,h] <- exp(alpha - m[dst,h]); atomic sum per (dst,h)
__global__ void hgt_expsum(const int* __restrict__ dst, float* __restrict__ alpha,
                           const float* __restrict__ mbuf, float* __restrict__ sbuf, int E) {
  int t = blockIdx.x * blockDim.x + threadIdx.x;
  if (t >= E * 8) return;
  const int e = t >> 3, h = t & 7;
  const int d = dst[e];
  const float v = expf(alpha[t] - mbuf[(size_t)d * 8 + h]);
  alpha[t] = v;
  atomicAdd(&sbuf[(size_t)d * 8 + h], v);
}

// out[dst, c] += v_rel[src, c] * alpha[e, c/16] / (s[dst, c/16] + eps)
__global__ void hgt_scatter(const int* __restrict__ src, const int* __restrict__ dst,
                            const float* __restrict__ alpha, const float* __restrict__ sbuf,
                            const float* __restrict__ vrel, float* __restrict__ out,
                            long long total) {
  long long t = (long long)blockIdx.x * blockDim.x + threadIdx.x;
  if (t >= total) return;
  const int e = (int)(t >> 7);
  const int c = (int)(t & 127);
  const int h = c >> 4;
  const int s = src[e], d = dst[e];
  const float coef = alpha[(size_t)e * 8 + h] / (sbuf[(size_t)d * 8 + h] + 1e-16f);
  atomicAdd(&out[(size_t)d * 128 + c], vrel[(size_t)s * 128 + c] * coef);
}

// ---------------------------------------------------------------------------
static inline int imax3(int a, int b, int c) { int m = a > b ? a : b; return m > c ? m : c; }

extern "C" void kernel_launch(void* const* d_in, const int* in_sizes, int n_in,
                              void* d_out, int out_size, void* d_ws, size_t ws_size,
                              hipStream_t stream) {
  const float* x_author = (const float*)d_in[0];
  const float* x_paper  = (const float*)d_in[1];
  const float* W_in_a = (const float*)d_in[2];  const float* b_in_a = (const float*)d_in[3];
  const float* W_in_p = (const float*)d_in[4];  const float* b_in_p = (const float*)d_in[5];
  const float* Wk_a = (const float*)d_in[6];   const float* bk_a = (const float*)d_in[7];
  const float* Wq_a = (const float*)d_in[8];   const float* bq_a = (const float*)d_in[9];
  const float* Wv_a = (const float*)d_in[10];  const float* bv_a = (const float*)d_in[11];
  const float* Wa_a = (const float*)d_in[12];  const float* ba_a = (const float*)d_in[13];
  const float* skip_a = (const float*)d_in[14];
  const float* Wk_p = (const float*)d_in[15];  const float* bk_p = (const float*)d_in[16];
  const float* Wq_p = (const float*)d_in[17];  const float* bq_p = (const float*)d_in[18];
  const float* Wv_p = (const float*)d_in[19];  const float* bv_p = (const float*)d_in[20];
  const float* Wa_p = (const float*)d_in[21];  const float* ba_p = (const float*)d_in[22];
  const float* skip_p = (const float*)d_in[23];
  const float* ar_w = (const float*)d_in[24]; const float* mr_w = (const float*)d_in[25]; const float* pr_w = (const float*)d_in[26];
  const float* ar_r = (const float*)d_in[27]; const float* mr_r = (const float*)d_in[28]; const float* pr_r = (const float*)d_in[29];
  const float* ar_c = (const float*)d_in[30]; const float* mr_c = (const float*)d_in[31]; const float* pr_c = (const float*)d_in[32];
  const int* src_w = (const int*)d_in[33]; const int* dst_w = (const int*)d_in[34];
  const int* src_r = (const int*)d_in[35]; const int* dst_r = (const int*)d_in[36];
  const int* src_c = (const int*)d_in[37]; const int* dst_c = (const int*)d_in[38];

  const int NA = in_sizes[0] / 64;     // 50000 (multiple of 16)
  const int NP = in_sizes[1] / 128;    // 100000 (multiple of 16)
  const int EW = in_sizes[33], ER = in_sizes[35], EC = in_sizes[37];
  const int Emax = imax3(EW, ER, EC);
  const int Nmax = NA > NP ? NA : NP;

  // ---- workspace carve (floats) ----
  float* ws = (float*)d_ws;
  size_t off = 0;
  auto carve = [&](size_t n) { float* p = ws + off; off += n; return p; };
  float* h_a = carve((size_t)NA * 128);
  float* h_p = carve((size_t)NP * 128);
  float* k_a = carve((size_t)NA * 128);
  float* q_a = carve((size_t)NA * 128);
  float* v_a = carve((size_t)NA * 128);
  float* k_p = carve((size_t)NP * 128);
  float* q_p = carve((size_t)NP * 128);
  float* v_p = carve((size_t)NP * 128);
  float* krel = carve((size_t)NP * 128);   // sized for largest source set
  float* vrel = carve((size_t)NP * 128);
  float* alpha = carve((size_t)Emax * 8);
  float* mbuf = carve((size_t)Nmax * 8);
  float* sbuf = carve((size_t)Nmax * 8);
  float* out_a = carve((size_t)NA * 128);
  float* out_p = carve((size_t)NP * 128);

  float* res_a = (float*)d_out;                       // [NA x 128]
  float* res_p = (float*)d_out + (size_t)NA * 128;    // [NP x 128]

  const dim3 B256(256);
  auto g1 = [](size_t n) { return dim3((unsigned)((n + 255) / 256)); };

  // ---- zero message accumulators ----
  hgt_fill<<<g1((size_t)NA * 128), B256, 0, stream>>>(out_a, 0.0f, (size_t)NA * 128);
  hgt_fill<<<g1((size_t)NP * 128), B256, 0, stream>>>(out_p, 0.0f, (size_t)NP * 128);

  // ---- input projections (relu) ----
  hgt_gemm128<<<dim3(NA / 16), B256, 0, stream>>>(x_author, W_in_a, b_in_a, h_a, 64, 1);
  hgt_gemm128<<<dim3(NP / 16), B256, 0, stream>>>(x_paper,  W_in_p, b_in_p, h_p, 128, 1);

  // ---- k/q/v projections ----
  hgt_gemm128<<<dim3(NA / 16), B256, 0, stream>>>(h_a, Wk_a, bk_a, k_a, 128, 0);
  hgt_gemm128<<<dim3(NA / 16), B256, 0, stream>>>(h_a, Wq_a, bq_a, q_a, 128, 0);
  hgt_gemm128<<<dim3(NA / 16), B256, 0, stream>>>(h_a, Wv_a, bv_a, v_a, 128, 0);
  hgt_gemm128<<<dim3(NP / 16), B256, 0, stream>>>(h_p, Wk_p, bk_p, k_p, 128, 0);
  hgt_gemm128<<<dim3(NP / 16), B256, 0, stream>>>(h_p, Wq_p, bq_p, q_p, 128, 0);
  hgt_gemm128<<<dim3(NP / 16), B256, 0, stream>>>(h_p, Wv_p, bv_p, v_p, 128, 0);

  // ---- relations (processed sequentially; krel/vrel/alpha/m/s reused) ----
  struct RelDesc {
    const float *ksrc, *vsrc, *qdst, *ar, *mr, *pr;
    const int *src, *dst;
    float* out; int E, n_src, n_dst;
  };
  const RelDesc rels[3] = {
    { k_a, v_a, q_p, ar_w, mr_w, pr_w, src_w, dst_w, out_p, EW, NA, NP },  // author -> paper
    { k_p, v_p, q_a, ar_r, mr_r, pr_r, src_r, dst_r, out_a, ER, NP, NA },  // paper  -> author
    { k_p, v_p, q_p, ar_c, mr_c, pr_c, src_c, dst_c, out_p, EC, NP, NP },  // paper  -> paper
  };
  for (int i = 0; i < 3; ++i) {
    const RelDesc& R = rels[i];
    hgt_rel_transform<<<dim3(R.n_src / 16), B256, 0, stream>>>(R.ksrc, R.ar, krel);
    hgt_rel_transform<<<dim3(R.n_src / 16), B256, 0, stream>>>(R.vsrc, R.mr, vrel);
    const size_t nh = (size_t)R.n_dst * 8;
    hgt_fill<<<g1(nh), B256, 0, stream>>>(mbuf, -3.402823466e38f, nh);
    hgt_fill<<<g1(nh), B256, 0, stream>>>(sbuf, 0.0f, nh);
    hgt_alpha<<<g1((size_t)R.E * 8), B256, 0, stream>>>(R.qdst, krel, R.src, R.dst,
                                                        R.pr, alpha, mbuf, R.E);
    hgt_expsum<<<g1((size_t)R.E * 8), B256, 0, stream>>>(R.dst, alpha, mbuf, sbuf, R.E);
    const long long tot = (long long)R.E * 128;
    hgt_scatter<<<g1((size_t)tot), B256, 0, stream>>>(R.src, R.dst, alpha, sbuf,
                                                      vrel, R.out, tot);
  }

  // ---- output projection + gated residual ----
  hgt_outproj<<<dim3(NA / 16), B256, 0, stream>>>(out_a, Wa_a, ba_a, h_a, skip_a, res_a);
  hgt_outproj<<<dim3(NP / 16), B256, 0, stream>>>(out_p, Wa_p, ba_p, h_p, skip_p, res_p);
}